// SymmetricContraction_8400956031126
// MI455X (gfx1250) — compile-verified
//
#include <hip/hip_runtime.h>
#include <cstdint>

typedef __attribute__((ext_vector_type(2))) float v2f;
typedef __attribute__((ext_vector_type(8))) float v8f;

#define NODES 2048
#define CH    128
#define KD    16
#define ROWS3 2304                 // sum_m m*16*16  (nu=3 rows)
#define NG    144                  // sum_m m*16     ((mi,k1) groups; out_mi = g>>4, k1 = g&15)
#define C3_PER_CH (ROWS3*KD)       // 36864
#define C2_PER_CH (NG*KD)          // 2304
#define C1_PER_CH (NG)             // 144
#define NC3 (CH*C3_PER_CH)
#define NC2 (CH*C2_PER_CH)
#define NC1 (CH*C1_PER_CH)
#define NPB 256                    // nodes per workgroup (4 waves x 4 subtiles x 16)
#define NSUB 4                     // node subtiles (WMMA column blocks) per wave

struct UW {
  const float* U1[3]; const float* U2[3]; const float* U3[3];
  const float* W1[3]; const float* W2[3]; const float* W3[3];
};

// ---------------- Kernel 1: fold W into U once (C = U . W), ~1% of FLOPs ---------------
__global__ __launch_bounds__(256) void mace_precompute(UW p,
    float* __restrict__ wsC3, float* __restrict__ wsC2, float* __restrict__ wsC1) {
  int gid = blockIdx.x * 256 + threadIdx.x;
  if (gid < NC3) {
    int c = gid / C3_PER_CH;
    int r = gid - c * C3_PER_CH;
    int row = r >> 4, i = r & 15;
    int blk, rl;
    if (row < 256)       { blk = 0; rl = row; }
    else if (row < 1024) { blk = 1; rl = row - 256; }
    else                 { blk = 2; rl = row - 1024; }
    const float* U = p.U3[blk];
    const float* W = p.W3[blk];
    float s = 0.f;
#pragma unroll
    for (int q = 0; q < 20; ++q) s += U[(rl * 16 + i) * 20 + q] * W[q * CH + c];
    wsC3[gid] = s;                                  // layout [c][row][i]
  } else if (gid < NC3 + NC2) {
    int t = gid - NC3;
    int c = t / C2_PER_CH;
    int r = t - c * C2_PER_CH;
    int g = r >> 4, k2 = r & 15;
    int blk, gl;
    if (g < 16)      { blk = 0; gl = g; }
    else if (g < 64) { blk = 1; gl = g - 16; }
    else             { blk = 2; gl = g - 64; }
    const float* U = p.U2[blk];
    const float* W = p.W2[blk];
    float s = 0.f;
#pragma unroll
    for (int q = 0; q < 4; ++q) s += U[(gl * 16 + k2) * 4 + q] * W[q * CH + c];
    wsC2[t] = s;                                    // layout [c][g][k2]
  } else if (gid < NC3 + NC2 + NC1) {
    int t = gid - NC3 - NC2;
    int c = t / C1_PER_CH;
    int r = t - c * C1_PER_CH;
    int mo = r >> 4, k1 = r & 15;
    int blk, lmi;
    if (mo < 1)      { blk = 0; lmi = 0; }
    else if (mo < 4) { blk = 1; lmi = mo - 1; }
    else             { blk = 2; lmi = mo - 4; }
    wsC1[t] = p.U1[blk][lmi * 16 + k1] * p.W1[blk][c];   // layout [c][g]
  }
}

// ---------------- CDNA5 async global->LDS staging ----------------
__device__ __forceinline__ void async_ld_b128(const float* gptr, const float* lptr) {
  // Generic LDS pointers carry the raw LDS byte offset in their low 32 bits.
  unsigned lds = (unsigned)(size_t)lptr;
  asm volatile("global_load_async_to_lds_b128 %0, %1, off"
               :: "v"(lds), "v"(gptr) : "memory");
}
__device__ __forceinline__ void wait_async0() {
  asm volatile("s_wait_asynccnt 0x0" ::: "memory");
}

// ---------------- Kernel 2: WMMA contraction + recursive peel ----------------
// Workgroup = (channel c, 256-node tile). Wave handles 4 subtiles of 16 nodes so
// each C3 A-operand tile (1KB from L2) feeds 16 chained WMMAs (32 FLOP/L2 byte).
// GEMM: D[k2(tile row), node] = sum_i C3[g*16+k2, i] * A[node, i]  via 4x chained
// v_wmma_f32_16x16x4_f32 (K=16). Then per lane (lane=node column, VGPR+half = k2):
//   b2 = sum_k2 (C2[g,k2] + D[k2]) * a[k2]   (cross-half via shfl_xor 16)
//   out[mi] += (C1[g] + b2) * a[k1]          (g = mi*16 + k1)
__global__ __launch_bounds__(128) void mace_contract(
    const float* __restrict__ A, const float* __restrict__ wsC3,
    const float* __restrict__ wsC2, const float* __restrict__ wsC1,
    float* __restrict__ out) {
  __shared__ float sA[NPB * KD];      // 16 KB: A[node,:] slice for this (tile, channel)
  __shared__ float sC2[C2_PER_CH];    //  9 KB
  __shared__ float sC1[C1_PER_CH];    // 576 B

  const int tid = threadIdx.x;
  const int c   = blockIdx.x & (CH - 1);
  const int nb  = (blockIdx.x >> 7) * NPB;

  // Stage A slice: 1024 float4 (node rows are 16 contiguous floats)
#pragma unroll
  for (int j = 0; j < 8; ++j) {
    int f  = j * 128 + tid;
    int nl = f >> 2, q = f & 3;
    async_ld_b128(A + ((size_t)(nb + nl) * CH + c) * KD + q * 4, &sA[f * 4]);
  }
  // Stage C2: 576 float4
  const float* gC2 = wsC2 + (size_t)c * C2_PER_CH;
#pragma unroll
  for (int j = 0; j < 5; ++j) {
    int z = j * 128 + tid;
    if (z < C2_PER_CH / 4) async_ld_b128(gC2 + z * 4, &sC2[z * 4]);
  }
  // Stage C1: 36 float4
  if (tid < C1_PER_CH / 4)
    async_ld_b128(wsC1 + (size_t)c * C1_PER_CH + tid * 4, &sC1[tid * 4]);

  wait_async0();
  __syncthreads();

  const int lane = tid & 31;
  const int wave = tid >> 5;
  const int col  = lane & 15;   // WMMA column = node within subtile
  const int hi   = lane >> 4;   // lane half selects K/M offsets per f32 layouts
  const float* gC3 = wsC3 + (size_t)c * C3_PER_CH;   // L2-resident channel slice

  int nl[NSUB];
#pragma unroll
  for (int t = 0; t < NSUB; ++t) nl[t] = wave * (16 * NSUB) + t * 16 + col;

  // B operands (A-matrix slice, 4x16 per chained step): lane: N=col, K = 2*hi + {0,1}
  v2f bop[NSUB][4];
#pragma unroll
  for (int t = 0; t < NSUB; ++t)
#pragma unroll
    for (int s = 0; s < 4; ++s) {
      int i0 = 4 * s + 2 * hi;
      bop[t][s] = v2f{sA[nl[t] * KD + i0], sA[nl[t] * KD + i0 + 1]};
    }

  float* outp[NSUB];
#pragma unroll
  for (int t = 0; t < NSUB; ++t)
    outp[t] = out + ((size_t)(nb + nl[t]) * CH + c) * 9;

  for (int mi = 0; mi < 9; ++mi) {
    float acc[NSUB];
#pragma unroll
    for (int t = 0; t < NSUB; ++t) acc[t] = 0.f;

    for (int k1 = 0; k1 < 16; ++k1) {
      const int g = mi * 16 + k1;
      // A operand: lane M=col -> C3 row g*16+col; K = 4*s + 2*hi + {0,1}
      const float* rowp = gC3 + (size_t)(g * 16 + col) * KD + 2 * hi;

      // Issue all 4 operand loads up front (distinct regs -> one s_clause,
      // staged s_wait_loadcnt across the 16-WMMA chain).
      v2f a4[4];
#pragma unroll
      for (int s = 0; s < 4; ++s) a4[s] = *(const v2f*)(rowp + 4 * s);

      v8f d[NSUB];
#pragma unroll
      for (int t = 0; t < NSUB; ++t) d[t] = v8f{0.f,0.f,0.f,0.f,0.f,0.f,0.f,0.f};
#pragma unroll
      for (int s = 0; s < 4; ++s) {
#pragma unroll
        for (int t = 0; t < NSUB; ++t)
          d[t] = __builtin_amdgcn_wmma_f32_16x16x4_f32(false, a4[s], false, bop[t][s],
                                                       (short)0, d[t], false, false);
      }
      // nu=2 peel: D VGPR v holds k2 = v + 8*hi for column `col`
      float part[NSUB];
#pragma unroll
      for (int t = 0; t < NSUB; ++t) part[t] = 0.f;
#pragma unroll
      for (int v = 0; v < 8; ++v) {
        int k2 = v + 8 * hi;
        float c2 = sC2[g * KD + k2];
#pragma unroll
        for (int t = 0; t < NSUB; ++t)
          part[t] += (d[t][v] + c2) * sA[nl[t] * KD + k2];
      }
      float c1 = sC1[g];
#pragma unroll
      for (int t = 0; t < NSUB; ++t) {
        float b2 = part[t] + __shfl_xor(part[t], 16, 32);
        acc[t] += (c1 + b2) * sA[nl[t] * KD + k1];   // nu=1 peel
      }
    }
    if (hi == 0) {
#pragma unroll
      for (int t = 0; t < NSUB; ++t) outp[t][mi] = acc[t];
    }
  }
}

extern "C" void kernel_launch(void* const* d_in, const int* in_sizes, int n_in,
                              void* d_out, int out_size, void* d_ws, size_t ws_size,
                              hipStream_t stream) {
  // d_in order: A, Us[m][nu] (row-major over m then nu), Ws[m][nu]
  UW p;
  p.U1[0] = (const float*)d_in[1];  p.U2[0] = (const float*)d_in[2];  p.U3[0] = (const float*)d_in[3];
  p.U1[1] = (const float*)d_in[4];  p.U2[1] = (const float*)d_in[5];  p.U3[1] = (const float*)d_in[6];
  p.U1[2] = (const float*)d_in[7];  p.U2[2] = (const float*)d_in[8];  p.U3[2] = (const float*)d_in[9];
  p.W1[0] = (const float*)d_in[10]; p.W2[0] = (const float*)d_in[11]; p.W3[0] = (const float*)d_in[12];
  p.W1[1] = (const float*)d_in[13]; p.W2[1] = (const float*)d_in[14]; p.W3[1] = (const float*)d_in[15];
  p.W1[2] = (const float*)d_in[16]; p.W2[2] = (const float*)d_in[17]; p.W3[2] = (const float*)d_in[18];

  float* ws   = (float*)d_ws;         // needs ~20.1 MB
  float* wsC3 = ws;
  float* wsC2 = ws + NC3;
  float* wsC1 = wsC2 + NC2;

  const int total = NC3 + NC2 + NC1;
  mace_precompute<<<(total + 255) / 256, 256, 0, stream>>>(p, wsC3, wsC2, wsC1);
  mace_contract<<<CH * (NODES / NPB), 128, 0, stream>>>(
      (const float*)d_in[0], wsC3, wsC2, wsC1, (float*)d_out);
}